// SABRBackwardModel_4578435138180
// MI455X (gfx1250) — compile-verified
//
#include <hip/hip_runtime.h>
#include <hip/hip_bf16.h>

// SABR implied-vol surface, specialized for BETA=1, R=Q=0, RHO=-0.7, VOLVOL=0.3.
// Memory-bound: ~210MB of output stores -> coalesced 128-bit stores dominate.
//
// Stage 1: each thread computes its element's 25 surface values into LDS in
//          the exact flat layout of this block's global output slice.
//          out[t][k] = B(t) * S[k]  (B affine in t; 8 logs+4 sqrts+9 divs/elem).
// Stage 2: block drains LDS -> global, 16B per lane, fully coalesced.
//          Preferred path: gfx1250 GLOBAL_STORE_ASYNC_FROM_LDS_B128 (direct
//          LDS->memory DMA, ASYNCcnt-tracked, no VGPR bounce).
//          Fallback: ds_load_b128 + global_store_b128 th:NT.

typedef float v4f __attribute__((ext_vector_type(4)));
typedef int   v4i __attribute__((ext_vector_type(4)));

// Builtin signature (from hipcc diagnostic): param0 is
//   'int __vector(4) __device__ *'  i.e. v4i in amdgpu addrspace(1).
typedef __attribute__((address_space(1))) v4i* gv4i_ptr;   // global (__device__)
typedef __attribute__((address_space(3))) v4i* lv4i_ptr;   // LDS (__shared__)

#define TPB 256                  // threads per block (8 wave32s)
#define OPE 25                   // outputs per element
#define BLK_FLOATS (TPB * OPE)   // 6400 floats = 25600 B of LDS
#define BLK_V4 (BLK_FLOATS / 4)  // 1600 float4s

#if __has_builtin(__builtin_amdgcn_global_store_async_from_lds_b128)
#define HAVE_ASYNC_STORE 1
#else
#define HAVE_ASYNC_STORE 0
#endif

__global__ __launch_bounds__(TPB)
void sabr_iv_kernel(const float* __restrict__ vol,
                    const float* __restrict__ price,
                    float* __restrict__ out)
{
    __shared__ float lds[BLK_FLOATS];

    const int tid = threadIdx.x;
    // 4096*512 = 2,097,152 elements, divisible by 256 -> exact grid, no guard.
    const long long e = (long long)blockIdx.x * TPB + tid;

    const float v = vol[e];
    const float p = price[e];

    // ---- Stage 1: per-element math ----------------------------------------
    //   B(t) = 1 + t * (-0.0525*v + 0.0019875)
    const float d = fmaf(-0.0525f, v, 0.0019875f);

    const float mon[5] = {0.7f, 0.85f, 1.0f, 1.15f, 1.3f};
    const float volvol_over_v = 0.3f / v;   // VOLVOL * x / v with x == 1

    float S[5];
#pragma unroll
    for (int k = 0; k < 5; ++k) {
        if (k == 2) {
            // F == K branch: v * B / F^(1-beta) = v * B  (beta == 1)
            S[k] = v;
        } else {
            const float K     = p * mon[k];
            const float logFK = __builtin_logf(p / K);      // log(F/K), F = p
            const float Phi   = volvol_over_v * logFK;
            // 1 - 2*RHO*Phi + Phi^2 = 1 + 1.4*Phi + Phi^2  (always > 0)
            const float rad   = fmaf(Phi, Phi, fmaf(1.4f, Phi, 1.0f));
            const float Chi   = __builtin_logf(
                (__builtin_sqrtf(rad) + Phi + 0.7f) * (1.0f / 1.7f));
            S[k] = v * Phi / (Chi + 1e-8f);   // A*B*Phi/(Chi+eps), A = v
        }
    }

    // 25 outputs = outer product B(t) x S[k]; LDS layout mirrors global layout
    float* myl = &lds[tid * OPE];   // stride 25 (odd) -> bank-conflict-free
#pragma unroll
    for (int t = 0; t < 5; ++t) {
        const float B = fmaf((float)t, d, 1.0f);
#pragma unroll
        for (int k = 0; k < 5; ++k) {
            myl[t * 5 + k] = B * S[k];
        }
    }

    __syncthreads();   // gfx1250 split barrier (signal/wait -1)

    // ---- Stage 2: coalesced drain, LDS -> global, 16B per lane ------------
    v4f* out4 = (v4f*)out + (size_t)blockIdx.x * BLK_V4;

#if HAVE_ASYNC_STORE
    // Direct LDS -> memory DMA (gfx1250): global_store_async_from_lds_b128.
    // Per lane: copies 16 bytes from LDS at lds[4*g] to global memory at out4[g]; tracked with ASYNCcnt.
#pragma unroll
    for (int g = tid; g < BLK_V4; g += TPB) {      // 7 rounds (last partial)
        __builtin_amdgcn_global_store_async_from_lds_b128(
            (gv4i_ptr)&out4[g], (lv4i_ptr)&lds[4 * g],
            /*offset=*/0, /*cpol=*/0);
    }
#if __has_builtin(__builtin_amdgcn_s_wait_asynccnt)
    __builtin_amdgcn_s_wait_asynccnt(0);
#else
    asm volatile("s_wait_asynccnt 0x0" ::: "memory");
#endif
#else
    // Fallback: LDS -> VGPR -> global (non-temporal, write-once output).
    const v4f* lds4 = (const v4f*)lds;
#pragma unroll
    for (int g = tid; g < BLK_V4; g += TPB) {      // 7 rounds (last partial)
        const v4f val = lds4[g];                    // ds_load_b128 (16B aligned)
        __builtin_nontemporal_store(val, &out4[g]); // global_store_b128 th:NT
    }
#endif
}

extern "C" void kernel_launch(void* const* d_in, const int* in_sizes, int n_in,
                              void* d_out, int out_size, void* d_ws, size_t ws_size,
                              hipStream_t stream) {
    const float* vol   = (const float*)d_in[0];
    const float* price = (const float*)d_in[1];
    float* out = (float*)d_out;

    const int n_elem = in_sizes[0];               // 4096*512 = 2,097,152
    const int blocks = (n_elem + TPB - 1) / TPB;  // 8192 (exact: n % 256 == 0)

    sabr_iv_kernel<<<blocks, TPB, 0, stream>>>(vol, price, out);
}